// TrainableModel_32719060861608
// MI455X (gfx1250) — compile-verified
//
#include <hip/hip_runtime.h>

typedef __attribute__((ext_vector_type(2))) float v2f;
typedef __attribute__((ext_vector_type(8))) float v8f;

#define NEV 65536
#define NOG 64
#define NXG 256
#define NBG 256

// constants (match reference float32 semantics)
#define kPI     3.14159265358979323846f
#define kG2     0.12f
#define kM2     0.8803f
#define kS      140.0f
#define kALPHA0 (1.0f / 137.035999f)
#define kME2    (0.000511f * 0.000511f)
#define kLXMIN  (-9.210340371976182f)   /* log(1e-4) */
#define kLBMIN  (-6.907755278982137f)   /* log(1e-3) */
#define kLBMAX  (3.9120230054281460f)   /* log(50)   */
#define kSXn    (255.0f / (0.0f - kLXMIN))
#define kSBn    (255.0f / (kLBMAX - kLBMIN))
#define kCLMP   254.9999f               /* NX-1-1e-4 */
#define kLOG2E  1.4426950408889634f

// fast 2^x (arg already in log2 domain, always finite & <= 0 here)
__device__ __forceinline__ float fast_exp2(float a)
{
#if __has_builtin(__builtin_amdgcn_exp2f)
    return __builtin_amdgcn_exp2f(a);       // v_exp_f32
#else
    return exp2f(a);
#endif
}

// ---------------------------------------------------------------------------
// Relayout (8,256,256) grids -> (256,256,8): the 8 flavors at a grid corner
// become contiguous (32B), and a (j0, j0+1) corner pair is one 64B region.
// ---------------------------------------------------------------------------
__global__ void __launch_bounds__(256)
transpose_grids(const float* __restrict__ pdfg, const float* __restrict__ ffg,
                float* __restrict__ pdfT, float* __restrict__ ffT)
{
    int idx = blockIdx.x * 256 + threadIdx.x;     // 0 .. 8*65536-1
    if (idx >= 8 * NXG * NBG) return;
    int f  = idx >> 16;
    int ij = idx & 0xFFFF;
    pdfT[ij * 8 + f] = pdfg[idx];
    ffT [ij * 8 + f] = ffg [idx];
}

// Load the 16 values at corners (i,j,f=0..7) and (i,j+1,f=0..7).
template<bool TRANS>
__device__ __forceinline__ void cornerPair(const float* __restrict__ g, int i, int j, float v[16])
{
    if (TRANS) {
        const float4* q = reinterpret_cast<const float4*>(g + ((((i << 8) + j)) << 3));
        float4 a = q[0], b = q[1], c = q[2], d = q[3];   // 4x global_load_b128, 1 base addr
        v[0]  = a.x; v[1]  = a.y; v[2]  = a.z; v[3]  = a.w;
        v[4]  = b.x; v[5]  = b.y; v[6]  = b.z; v[7]  = b.w;
        v[8]  = c.x; v[9]  = c.y; v[10] = c.z; v[11] = c.w;
        v[12] = d.x; v[13] = d.y; v[14] = d.z; v[15] = d.w;
    } else {
        int base = (i << 8) + j;
#pragma unroll
        for (int f = 0; f < 8; ++f) {
            v[f]     = g[f * (NXG * NBG) + base];
            v[8 + f] = g[f * (NXG * NBG) + base + 1];
        }
    }
}

// ---------------------------------------------------------------------------
// One wave handles 16 events x 64 Ogata points. Lane L (L<16) computes event
// L with k-slots {4s,4s+1}; lane L+16 the same event with k-slots {4s+2,4s+3}
// (== the f32 16x16x4 WMMA A-fragment layout). B = all-ones, so each
// V_WMMA_F32_16X16X4_F32 accumulates exact f32 row sums of the weighted,
// pre-scaled integrand; after 16 steps the C fragment holds the outputs.
// ---------------------------------------------------------------------------
template<bool TRANS>
__global__ void __launch_bounds__(256)
sidis_fuut_kernel(const float* __restrict__ ev,
                  const float* __restrict__ pdfg,   // transposed if TRANS
                  const float* __restrict__ ffg,    // transposed if TRANS
                  const float* __restrict__ ogx,
                  const float* __restrict__ ogw,
                  const float* __restrict__ fnp,
                  float* __restrict__ out)
{
    __shared__ float4 s_tab[NOG];                 // (log ogx, ogx^2, w, 0)
    const int t = threadIdx.x;
    if (t < NOG) {
        float u = ogx[t];
        s_tab[t] = make_float4(logf(u), u * u, ogw[t], 0.0f);
    }
    __syncthreads();

    const int lane  = t & 31;
    const int wave  = t >> 5;
    const int tile  = blockIdx.x * 8 + wave;       // 0 .. 4095
    const int m     = lane & 15;                   // event-in-tile == WMMA row M
    const int hi    = lane >> 4;                   // k-slot group
    const int n     = tile * 16 + m;

    const float4 e4 = *reinterpret_cast<const float4*>(ev + 4 * n);
    const float x = e4.x, PhT = e4.y, Q = e4.z, z = e4.w;

    const float qT     = PhT / z;
    const float Q2     = Q * Q;
    const float lx     = __logf(x);
    const float lz     = __logf(z);
    const float lqT    = __logf(qT);
    const float invqT2 = 1.0f / (qT * qT);

    // x-direction grid coordinates (shared across all k for this event)
    float fx  = fminf(fmaxf((lx - kLXMIN) * kSXn, 0.0f), kCLMP);
    int   i0  = (int)fx;  const float tx = fx - (float)i0;
    float fz  = fminf(fmaxf((lz - kLXMIN) * kSXn, 0.0f), kCLMP);
    int   iz0 = (int)fz;  const float tz = fz - (float)iz0;

    // fnp-derived scalars (uniform scalar loads)
    const float lam_p = __logf(1.0f + __expf(fnp[0]));      // softplus
    const float lam_f = __logf(1.0f + __expf(fnp[1]));
    const float sig2  = 1.0f / (1.0f + __expf(-fnp[2]));    // sigmoid
    const float sig3  = 1.0f / (1.0f + __expf(-fnp[3]));

    // fused Gaussian: evo2*f_pdf*f_ff = exp(-Cc*bT^2) = 2^(-Cc*log2e*bT^2)
    const float Cc = kG2 * __logf(Q2)
                   + lam_p * (1.0f - sig2 * lx)
                   + lam_f * (1.0f + sig3) / (z * z);
    const float Ccl2 = Cc * kLOG2E;

    // per-event prefactor, folded with 1/qT^2
    const float alpha = kALPHA0 / (1.0f - kALPHA0 * (1.0f / (3.0f * kPI)) * __logf(Q2 / kME2));
    const float gam   = 2.0f * kM2 * x / Q;
    const float y     = Q2 / x * (1.0f / (kS - kM2));
    const float gy    = 0.25f * gam * gam * y * y;
    const float eps   = (1.0f - y - gy) / (1.0f - y + 0.5f * y * y + gy);
    const float pre   = 8.0f * kPI * kPI * alpha * alpha * z * z * qT / x / (Q2 * Q)
                      * y * y * 0.5f / (1.0f - eps) * (1.0f + gam * gam / (2.0f * x));
    const float scale = pre * invqT2;

    const float kE2[8] = {4.0f/9.0f, 1.0f/9.0f, 1.0f/9.0f, 4.0f/9.0f,
                          4.0f/9.0f, 1.0f/9.0f, 1.0f/9.0f, 4.0f/9.0f};

    auto integ = [&](int k) -> float {
        const float4 tab = s_tab[k];              // one ds_load_b128
        const float lb = tab.x - lqT;
        float fb = fminf(fmaxf((lb - kLBMIN) * kSBn, 0.0f), kCLMP);
        int   j0 = (int)fb;  const float tb = fb - (float)j0;

        float PA[16], PB[16], FA[16], FB[16];     // [f]=j0, [8+f]=j0+1
        cornerPair<TRANS>(pdfg, i0,      j0, PA);
        cornerPair<TRANS>(pdfg, i0 + 1,  j0, PB);
        cornerPair<TRANS>(ffg,  iz0,     j0, FA);
        cornerPair<TRANS>(ffg,  iz0 + 1, j0, FB);

        float S = 0.0f;
#pragma unroll
        for (int f = 0; f < 8; ++f) {
            float p0 = fmaf(tb, PA[8 + f] - PA[f], PA[f]);
            float p1 = fmaf(tb, PB[8 + f] - PB[f], PB[f]);
            float pv = fmaf(tx, p1 - p0, p0);
            float q0 = fmaf(tb, FA[8 + f] - FA[f], FA[f]);
            float q1 = fmaf(tb, FB[8 + f] - FB[f], FB[f]);
            float fv = fmaf(tz, q1 - q0, q0);
            S = fmaf(kE2[f], pv * fv, S);
        }
        const float bT2 = tab.y * invqT2;
        const float g   = fast_exp2(-Ccl2 * bT2); // v_mul + v_exp_f32
        return S * g * (tab.z * scale);
    };

    v8f c = {};
#pragma unroll 4
    for (int s = 0; s < 16; ++s) {
        const int k0 = 4 * s + 2 * hi;
        v2f A;  A[0] = integ(k0);  A[1] = integ(k0 + 1);   // K-slots per A-fragment layout
        v2f B;  B[0] = 1.0f;       B[1] = 1.0f;            // all-ones B -> exact row sums
        c = __builtin_amdgcn_wmma_f32_16x16x4_f32(
                /*neg_a=*/false, A, /*neg_b=*/false, B,
                /*c_mod=*/(short)0, c, /*reuse_a=*/false, /*reuse_b=*/false);
    }

    // C-fragment layout: VGPR r, lanes 0-15 -> M=r; lanes 16-31 -> M=8+r.
    const int ebase = tile * 16;
    if (lane == 0) {
        float4 lo = make_float4(c[0], c[1], c[2], c[3]);
        float4 hv = make_float4(c[4], c[5], c[6], c[7]);
        *reinterpret_cast<float4*>(out + ebase)     = lo;   // global_store_b128
        *reinterpret_cast<float4*>(out + ebase + 4) = hv;
    } else if (lane == 16) {
        float4 lo = make_float4(c[0], c[1], c[2], c[3]);
        float4 hv = make_float4(c[4], c[5], c[6], c[7]);
        *reinterpret_cast<float4*>(out + ebase + 8)  = lo;
        *reinterpret_cast<float4*>(out + ebase + 12) = hv;
    }
}

// ---------------------------------------------------------------------------
extern "C" void kernel_launch(void* const* d_in, const int* in_sizes, int n_in,
                              void* d_out, int out_size, void* d_ws, size_t ws_size,
                              hipStream_t stream)
{
    const float* ev   = (const float*)d_in[0];   // (65536,4)
    const float* pdfg = (const float*)d_in[1];   // (8,256,256)
    const float* ffg  = (const float*)d_in[2];   // (8,256,256)
    const float* ogx  = (const float*)d_in[3];   // (64,)
    const float* ogw  = (const float*)d_in[4];   // (64,)
    const float* fnp  = (const float*)d_in[5];   // (4,)
    float* out        = (float*)d_out;           // (65536,)

    const size_t gridElems = (size_t)8 * NXG * NBG;          // 524288
    const size_t needWS    = 2 * gridElems * sizeof(float);  // 4 MB

    if (ws_size >= needWS) {
        float* pdfT = (float*)d_ws;
        float* ffT  = pdfT + gridElems;
        transpose_grids<<<(int)((gridElems + 255) / 256), 256, 0, stream>>>(pdfg, ffg, pdfT, ffT);
        sidis_fuut_kernel<true><<<NEV / 16 / 8, 256, 0, stream>>>(ev, pdfT, ffT, ogx, ogw, fnp, out);
    } else {
        sidis_fuut_kernel<false><<<NEV / 16 / 8, 256, 0, stream>>>(ev, pdfg, ffg, ogx, ogw, fnp, out);
    }
}